// LatticeCNN_6133213298762
// MI455X (gfx1250) — compile-verified
//
#include <hip/hip_runtime.h>

typedef float v2f __attribute__((ext_vector_type(2)));
typedef float v8f __attribute__((ext_vector_type(8)));

#define S_DIM 32
#define G_OUT 128
#define BATCH_N 64
#define KB 16
#define ASTRIDE (KB + 2)   // 18 floats: rows 8B-aligned, krow even -> aligned v2f
#define BSTRIDE (KB + 2)

// ---- CDNA5 async global->LDS path (guarded; falls back to sync copies) ----
#if defined(__has_builtin)
#if __has_builtin(__builtin_amdgcn_global_load_async_to_lds_b32)
#define USE_ASYNC 1
#endif
#endif
#ifndef USE_ASYNC
#define USE_ASYNC 0
#endif

typedef __attribute__((address_space(1))) int as1_int;  // global ("__device__")
typedef __attribute__((address_space(3))) int as3_int;  // LDS ("__shared__")

__device__ __forceinline__ void copy_b32_to_lds(const float* g, float* l) {
#if USE_ASYNC
  // generic->AS1: global generic VA == global VA; generic->AS3: low 32 bits
  // of a generic shared VA are the LDS byte offset (aperture truncation).
  __builtin_amdgcn_global_load_async_to_lds_b32(
      (as1_int*)(uintptr_t)g, (as3_int*)(unsigned int)(uintptr_t)l, 0, 0);
#else
  *l = *g;
#endif
}

__device__ __forceinline__ void wait_async_lds() {
#if USE_ASYNC
#if __has_builtin(__builtin_amdgcn_s_wait_asynccnt)
  __builtin_amdgcn_s_wait_asynccnt(0);
#else
  asm volatile("s_wait_asynccnt 0" ::: "memory");
#endif
#endif
}

// Reduce dense 0/1 lattice tensors c[i, x, a] (32x32x8) to per-(x,a) source
// index tables: idx = the unique i with c!=0, or -1 if none (join edge).
__global__ void lattice_prep_tables(const float* __restrict__ cmx,
                                    const float* __restrict__ cmy,
                                    const float* __restrict__ cjx,
                                    const float* __restrict__ cjy,
                                    int* __restrict__ tab) {
  const int tid = threadIdx.x;  // 0..255 == x*8 + a
  const float* srcs[4] = {cmx, cmy, cjx, cjy};
  for (int t = 0; t < 4; ++t) {
    const float* c = srcs[t];
    int idx = -1;
    for (int i = 0; i < S_DIM; ++i)
      if (c[i * 256 + tid] != 0.0f) idx = i;
    tab[t * 256 + tid] = idx;
  }
}

// One block per spatial position (px,py): both lattice convs as implicit
// GEMMs (M=64 batch, N=128 feat, K=64*F) in fp32 WMMA, double-buffered
// async-to-LDS weight staging, fused bias + blend + leaky ReLU epilogue.
template <int F>
__global__ __launch_bounds__(256) void lattice_layer(
    const float* __restrict__ X,   // [64][F][32][32]
    const float* __restrict__ mw,  // [k][128], k = (a*8+b)*F + f
    const float* __restrict__ mb,  // [128]
    const float* __restrict__ jw,
    const float* __restrict__ jb,
    const int* __restrict__ tab,   // 4 tables of 256: mIx, mIy, jIx, jIy
    float* __restrict__ Y) {       // [64][128][32][32]
  __shared__ float Am[2][BATCH_N][ASTRIDE];
  __shared__ float Aj[2][BATCH_N][ASTRIDE];
  __shared__ float Bm[2][G_OUT][BSTRIDE];  // transposed: [g][k]
  __shared__ float Bj[2][G_OUT][BSTRIDE];

  const int pos = blockIdx.x;
  const int px = pos >> 5;
  const int py = pos & 31;
  const int tid = threadIdx.x;
  const int lane = tid & 31;
  const int wave = tid >> 5;       // 0..7
  const int mt = wave >> 1;        // m-tile 0..3 (16 rows each)
  const int ntb = (wave & 1) * 4;  // first of 4 n-tiles (16 cols each)
  const int lhi = lane >> 4;       // half-wave select
  const int llo = lane & 15;

  const int* mIx = tab + 0 * 256 + px * 8;
  const int* mIy = tab + 1 * 256 + py * 8;
  const int* jIx = tab + 2 * 256 + px * 8;
  const int* jIy = tab + 3 * 256 + py * 8;

  v8f zero = {};
  v8f accM[4], accJ[4];
#pragma unroll
  for (int n = 0; n < 4; ++n) { accM[n] = zero; accJ[n] = zero; }

  const int Ktot = 64 * F;
  const int nch = Ktot / KB;

  auto stage = [&](int k0, int buf) {
    // A tiles (gathered X with join-edge zero mask): sync stores, 4/thread.
#pragma unroll
    for (int e = 0; e < 4; ++e) {
      const int idx = e * 256 + tid;
      const int mm = idx >> 4;  // batch row
      const int t = idx & 15;   // k within chunk
      const int k = k0 + t;
      const int ab = k / F;
      const int f = k - ab * F;
      const int a = ab >> 3;
      const int b = ab & 7;
      const int im_x = mIx[a], im_y = mIy[b];
      const int ij_x = jIx[a], ij_y = jIy[b];
      Am[buf][mm][t] = (im_x >= 0 && im_y >= 0)
          ? X[((mm * F + f) * S_DIM + im_x) * S_DIM + im_y] : 0.0f;
      Aj[buf][mm][t] = (ij_x >= 0 && ij_y >= 0)
          ? X[((mm * F + f) * S_DIM + ij_x) * S_DIM + ij_y] : 0.0f;
    }
    // B tiles (weights, contiguous): async global->LDS into transposed tile.
#pragma unroll
    for (int e = 0; e < 8; ++e) {
      const int idx = e * 256 + tid;
      const int t = idx >> 7;   // k within chunk
      const int g = idx & 127;
      copy_b32_to_lds(mw + (k0 + t) * G_OUT + g, &Bm[buf][g][t]);
      copy_b32_to_lds(jw + (k0 + t) * G_OUT + g, &Bj[buf][g][t]);
    }
  };

  // prologue: stage chunk 0
  stage(0, 0);
  wait_async_lds();
  __syncthreads();

  for (int c = 0; c < nch; ++c) {
    const int buf = c & 1;
    // overlap: issue staging of chunk c+1 into the other buffer, then compute
    if (c + 1 < nch) stage((c + 1) * KB, buf ^ 1);

#pragma unroll
    for (int kk = 0; kk < KB; kk += 4) {
      const int krow = kk + (lhi << 1);  // even -> aligned v2f in LDS
      const int arow = mt * 16 + llo;
      const v2f am = *reinterpret_cast<const v2f*>(&Am[buf][arow][krow]);
      const v2f aj = *reinterpret_cast<const v2f*>(&Aj[buf][arow][krow]);
#pragma unroll
      for (int n = 0; n < 4; ++n) {
        const int col = (ntb + n) * 16 + llo;
        const v2f bm = *reinterpret_cast<const v2f*>(&Bm[buf][col][krow]);
        const v2f bj = *reinterpret_cast<const v2f*>(&Bj[buf][col][krow]);
        accM[n] = __builtin_amdgcn_wmma_f32_16x16x4_f32(
            false, am, false, bm, (short)0, accM[n], false, false);
        accJ[n] = __builtin_amdgcn_wmma_f32_16x16x4_f32(
            false, aj, false, bj, (short)0, accJ[n], false, false);
      }
    }

    wait_async_lds();   // own async writes to buf^1 complete
    __syncthreads();    // all waves: reads of buf done, buf^1 fully staged
  }

  // fused epilogue: bias, 0.5/0.5 blend, leaky ReLU, scatter store
#pragma unroll
  for (int n = 0; n < 4; ++n) {
    const int g = (ntb + n) * 16 + llo;
    const float bm_ = mb[g];
    const float bj_ = jb[g];
#pragma unroll
    for (int r = 0; r < 8; ++r) {
      const int m = mt * 16 + (lhi << 3) + r;  // C/D layout: M = r + 8*lhi
      float v = 0.5f * (accM[n][r] + bm_) + 0.5f * (accJ[n][r] + bj_);
      v = v > 0.0f ? v : 0.01f * v;
      Y[((m * G_OUT + g) * S_DIM + px) * S_DIM + py] = v;
    }
  }
}

extern "C" void kernel_launch(void* const* d_in, const int* in_sizes, int n_in,
                              void* d_out, int out_size, void* d_ws, size_t ws_size,
                              hipStream_t stream) {
  (void)in_sizes; (void)n_in; (void)out_size; (void)ws_size;
  const float* x   = (const float*)d_in[0];
  const float* cmx = (const float*)d_in[1];
  const float* cmy = (const float*)d_in[2];
  const float* cjx = (const float*)d_in[3];
  const float* cjy = (const float*)d_in[4];
  const float* mw0 = (const float*)d_in[5];
  const float* mb0 = (const float*)d_in[6];
  const float* jw0 = (const float*)d_in[7];
  const float* jb0 = (const float*)d_in[8];
  const float* mw1 = (const float*)d_in[9];
  const float* mb1 = (const float*)d_in[10];
  const float* jw1 = (const float*)d_in[11];
  const float* jb1 = (const float*)d_in[12];

  int*   tab = (int*)d_ws;                    // 4 * 256 ints
  float* h   = (float*)((char*)d_ws + 4096);  // [64][128][32][32] fp32

  lattice_prep_tables<<<1, 256, 0, stream>>>(cmx, cmy, cjx, cjy, tab);

  lattice_layer<64><<<1024, 256, 0, stream>>>(x, mw0, mb0, jw0, jb0, tab, h);
  lattice_layer<128><<<1024, 256, 0, stream>>>(h, mw1, mb1, jw1, jb1, tab,
                                               (float*)d_out);
}